// Pool_56676388438709
// MI455X (gfx1250) — compile-verified
//
#include <hip/hip_runtime.h>

#define C_CH 64

typedef __attribute__((ext_vector_type(2))) float v2f;
typedef __attribute__((ext_vector_type(8))) float v8f;

// Phase 1: gather last-scale features per point and atomically scatter-add into
// the current-scale voxel accumulator (d_out), counting points per voxel.
// One wave32 per point: lane L handles channels 2L, 2L+1.
__global__ void scatter_add_kernel(const float* __restrict__ feat,
                                   const int* __restrict__ last,
                                   const int* __restrict__ cur,
                                   float* __restrict__ sums,
                                   unsigned* __restrict__ counts,
                                   int npts) {
    int tid   = blockIdx.x * blockDim.x + threadIdx.x;
    int lane  = threadIdx.x & 31;
    int wave  = tid >> 5;
    int nwaves = (gridDim.x * blockDim.x) >> 5;
    for (int p = wave; p < npts; p += nwaves) {
        int vl = last[p];                 // wave-uniform -> L0 hit across lanes
        int vc = cur[p];
        const float2* src = (const float2*)(feat + (size_t)vl * C_CH);
        float2 v = src[lane];             // 256B coalesced per wave
        float* dst = sums + (size_t)vc * C_CH + (lane << 1);
        atomicAdd(dst,     v.x);          // global_atomic_add_f32, L2-resident
        atomicAdd(dst + 1, v.y);
        if (lane == 0) atomicAdd(counts + vc, 1u);
    }
}

// Phase 2: out = diag(1/max(count,1)) * sums, one 16x16 f32 tile per wave via
// 4x V_WMMA_F32_16X16X4_F32 (K=4 slices of the 16-wide diagonal in A).
// Exact in f32: every non-diagonal A entry is 0.0f.
//
// Layouts (ISA 7.12.2, 32-bit):
//   A 16x4 : lanes 0-15 -> M=lane, reg0=K0, reg1=K1 ; lanes 16-31 -> M=lane-16, reg0=K2, reg1=K3
//   B 4x16 : reg0 holds rows K0 (lanes 0-15) / K2 (lanes 16-31); reg1 rows K1/K3; N = lane%16
//   C/D    : reg r -> row r (lanes 0-15) / row r+8 (lanes 16-31); N = lane%16
__global__ void finalize_wmma_kernel(float* __restrict__ out,
                                     const unsigned* __restrict__ counts,
                                     int ntiles) {
    int tid  = blockIdx.x * blockDim.x + threadIdx.x;
    int lane = threadIdx.x & 31;
    int wave = tid >> 5;
    if (wave >= ntiles) return;           // wave-uniform: EXEC stays all-ones

    int vt = wave >> 2;                   // C_CH/16 == 4 channel tiles
    int ct = wave & 3;
    int v0 = vt << 4;
    int c0 = ct << 4;
    int m  = lane & 15;
    int hi = (lane >> 4) & 1;             // lane half selects K pair
    int kb = hi << 1;                     // reg0 carries K=kb, reg1 K=kb+1

    unsigned cnt = counts[v0 + m];
    float inv = 1.0f / (float)(cnt ? cnt : 1u);

    v8f acc = {};
#pragma unroll
    for (int g = 0; g < 4; ++g) {
        v2f b;
        b[0] = out[(size_t)(v0 + 4 * g + kb)     * C_CH + c0 + m];
        b[1] = out[(size_t)(v0 + 4 * g + kb + 1) * C_CH + c0 + m];
        v2f a;
        a[0] = (m - 4 * g == kb)     ? inv : 0.0f;
        a[1] = (m - 4 * g == kb + 1) ? inv : 0.0f;
        acc = __builtin_amdgcn_wmma_f32_16x16x4_f32(
            /*neg_a=*/false, a, /*neg_b=*/false, b,
            /*c_mod=*/(short)0, acc, /*reuse_a=*/false, /*reuse_b=*/false);
    }

#pragma unroll
    for (int r = 0; r < 8; ++r) {
        int row = r + (hi << 3);
        out[(size_t)(v0 + row) * C_CH + c0 + m] = acc[r];
    }
}

extern "C" void kernel_launch(void* const* d_in, const int* in_sizes, int n_in,
                              void* d_out, int out_size, void* d_ws, size_t ws_size,
                              hipStream_t stream) {
    const float* feat = (const float*)d_in[0];
    const int*   last = (const int*)d_in[1];
    const int*   cur  = (const int*)d_in[2];
    float*       out  = (float*)d_out;
    unsigned*    counts = (unsigned*)d_ws;   // M * 4 bytes of scratch

    int npts = in_sizes[1];                  // N_PTS = 2,000,000
    int M    = out_size / C_CH;              // 131072 voxels

    // Zero the accumulator and counts (graph-capture-safe async memsets).
    hipMemsetAsync(d_out, 0, (size_t)out_size * sizeof(float), stream);
    hipMemsetAsync(d_ws,  0, (size_t)M * sizeof(unsigned), stream);

    const int threads = 256;                 // 8 wave32 per block
    const int wpb = threads / 32;

    int blocks1 = (npts + wpb - 1) / wpb;
    if (blocks1 > 131072) blocks1 = 131072;  // grid-stride covers the rest
    scatter_add_kernel<<<blocks1, threads, 0, stream>>>(feat, last, cur, out,
                                                        counts, npts);

    int ntiles  = (M / 16) * (C_CH / 16);    // 32768 tiles, one wave each
    int blocks2 = (ntiles + wpb - 1) / wpb;
    finalize_wmma_kernel<<<blocks2, threads, 0, stream>>>(out, counts, ntiles);
}